// PagedTransformerBlock_55233279427201
// MI455X (gfx1250) — compile-verified
//
#include <hip/hip_runtime.h>
#include <hip/hip_bf16.h>
#include <math.h>

// ---------------------------------------------------------------------------
// Problem constants (from reference)
// ---------------------------------------------------------------------------
#define DIM      2048
#define NH       16
#define HD       128
#define BATCH    32
#define BS       16
#define BLK_PER  64
#define MAXS     1024
#define HID      8192   // 4*DIM

typedef __attribute__((ext_vector_type(2))) float v2f;
typedef __attribute__((ext_vector_type(4))) float v4f;
typedef __attribute__((ext_vector_type(8))) float v8f;

// ---------------------------------------------------------------------------
// RMSNorm: out[b,:] = x[b,:] * rsqrt(mean(x^2)+eps) * w
// ---------------------------------------------------------------------------
__global__ void rmsnorm_kernel(const float* __restrict__ x,
                               const float* __restrict__ w,
                               float* __restrict__ out)
{
    const int b   = blockIdx.x;
    const int tid = threadIdx.x;               // 256 threads
    __shared__ float red[8];

    float ss = 0.f;
    for (int i = tid; i < DIM; i += 256) {
        float v = x[b * DIM + i];
        ss += v * v;
    }
    for (int off = 16; off > 0; off >>= 1) ss += __shfl_xor(ss, off, 32);
    if ((tid & 31) == 0) red[tid >> 5] = ss;
    __syncthreads();
    float tot = 0.f;
#pragma unroll
    for (int i = 0; i < 8; ++i) tot += red[i];
    const float rs = rsqrtf(tot * (1.0f / DIM) + 1e-5f);

    for (int i = tid; i < DIM; i += 256)
        out[b * DIM + i] = x[b * DIM + i] * rs * w[i];
}

// ---------------------------------------------------------------------------
// Split-K GEMM partial: P[y][z] = A[32, kz] @ W[N, kz]^T  over K-chunk z.
// blockIdx.y selects among up to 3 weight matrices (fused QKV / w1,w3).
// blockIdx.z = K-split index; chunk length = kchunk.
// Workgroup: 256 threads = 8 waves; tile = 32(M) x 64(N); K staged by 32.
// Each wave computes one 16x16 tile via V_WMMA_F32_16X16X4_F32.
// Weights are stream-once -> non-temporal loads.
// ---------------------------------------------------------------------------
__global__ void gemm32_part_kernel(const float* __restrict__ A,
                                   const float* W0, const float* W1, const float* W2,
                                   float* __restrict__ P,
                                   int K, int N, int kchunk)
{
    const float* W = (blockIdx.y == 0) ? W0 : (blockIdx.y == 1 ? W1 : W2);
    const int    S = gridDim.z;
    float*       C = P + (size_t)(blockIdx.y * S + blockIdx.z) * (size_t)(32 * N);

    __shared__ float As[32][36];   // padded stride 36: 16B aligned, conflict-free
    __shared__ float Ws[64][36];

    const int tid   = threadIdx.x;
    const int lane  = tid & 31;
    const int wid   = tid >> 5;        // 0..7
    const int mtile = wid & 1;         // 0..1  (M rows 0..15 / 16..31)
    const int ntl   = wid >> 1;        // 0..3  (N sub-tile within 64)
    const int n0    = blockIdx.x * 64;

    const int l    = lane & 15;        // 0..15
    const int half = lane >> 4;        // 0..1

    // cooperative-load indices
    const int ar = tid >> 3;           // 0..31
    const int ac = (tid & 7) * 4;      // 0,4,...,28

    const int kbeg = blockIdx.z * kchunk;
    const int kend = kbeg + kchunk;

    v8f acc = {};                      // 16x16 f32 accumulator (8 VGPRs)

    for (int k0 = kbeg; k0 < kend; k0 += 32) {
        // ---- stage A chunk [32 x 32] (coalesced 128B rows) ----
        const v4f av = *(const v4f*)(A + (size_t)ar * K + k0 + ac);
        *(v4f*)(&As[ar][ac]) = av;
        // ---- stage W chunk [64 x 32], stream-once -> non-temporal ----
        const v4f w0 = __builtin_nontemporal_load(
            (const v4f*)(W + (size_t)(n0 + ar)      * K + k0 + ac));
        const v4f w1 = __builtin_nontemporal_load(
            (const v4f*)(W + (size_t)(n0 + ar + 32) * K + k0 + ac));
        *(v4f*)(&Ws[ar][ac])      = w0;
        *(v4f*)(&Ws[ar + 32][ac]) = w1;
        // prefetch next weight chunk (global_prefetch_b8)
        if (k0 + 32 < kend) {
            __builtin_prefetch(W + (size_t)(n0 + ar)      * K + k0 + 32 + ac, 0, 1);
            __builtin_prefetch(W + (size_t)(n0 + ar + 32) * K + k0 + 32 + ac, 0, 1);
        }
        __syncthreads();

        // ---- 8 WMMA steps of K=4 each ----
        const float* Arow = &As[mtile * 16 + l][0];
        const float* Wrow = &Ws[ntl   * 16 + l][0];
#pragma unroll
        for (int kk = 0; kk < 32; kk += 4) {
            // f32 16x16x4 fragment layout:
            //   A: VGPR0 = K(2*half), VGPR1 = K(2*half+1), M = lane&15
            //   B: VGPR0 = K(2*half), VGPR1 = K(2*half+1), N = lane&15
            v2f a, b;
            a.x = Arow[kk + 2 * half];
            a.y = Arow[kk + 2 * half + 1];
            b.x = Wrow[kk + 2 * half];
            b.y = Wrow[kk + 2 * half + 1];
            acc = __builtin_amdgcn_wmma_f32_16x16x4_f32(
                false, a, false, b, (short)0, acc, false, false);
        }
        __syncthreads();
    }

    // ---- store partials: VGPR r -> (M = mtile*16 + 8*half + r, N = n0+ntl*16+l)
    const int col = n0 + ntl * 16 + l;
#pragma unroll
    for (int r = 0; r < 8; ++r) {
        const int row = mtile * 16 + 8 * half + r;
        C[(size_t)row * N + col] = acc[r];
    }
}

// ---------------------------------------------------------------------------
// Split-K reduction: O[y][i] = sum_z P[y][z][i] (+ residual[i])
// Fixed summation order -> bitwise deterministic.
// ---------------------------------------------------------------------------
__global__ void reduce_kernel(const float* __restrict__ P, int S, int n,
                              float* O0, float* O1, float* O2,
                              const float* __restrict__ residual)
{
    float* O = (blockIdx.y == 0) ? O0 : (blockIdx.y == 1 ? O1 : O2);
    const float* Py = P + (size_t)blockIdx.y * S * (size_t)n;
    const int i = blockIdx.x * 256 + threadIdx.x;
    if (i < n) {
        float s = 0.f;
        for (int z = 0; z < S; ++z) s += Py[(size_t)z * n + i];
        if (residual) s += residual[i];
        O[i] = s;
    }
}

// ---------------------------------------------------------------------------
// Fused split-K reduce + SwiGLU:  g = silu(sum_z Pa) * (sum_z Pc)
// Pa at P[0..S), Pc at P[S..2S)
// ---------------------------------------------------------------------------
__global__ void swiglu_reduce_kernel(const float* __restrict__ P, int S, int n,
                                     float* __restrict__ g)
{
    const int i = blockIdx.x * 256 + threadIdx.x;
    if (i < n) {
        float a = 0.f, c = 0.f;
        for (int z = 0; z < S; ++z) {
            a += P[(size_t)z * n + i];
            c += P[(size_t)(S + z) * n + i];
        }
        g[i] = a * (1.0f / (1.0f + __expf(-a))) * c;
    }
}

// ---------------------------------------------------------------------------
// Paged attention, one workgroup per (b, h). The freshly projected k/v row
// replaces the heap entry at position ctx-1 (== slot_mapping target; block
// tables give each sequence disjoint physical blocks).
// KV is stream-once -> non-temporal loads.
// ---------------------------------------------------------------------------
__global__ void paged_attn_kernel(const float* __restrict__ q,
                                  const float* __restrict__ knew,
                                  const float* __restrict__ vnew,
                                  const float* __restrict__ kheap,
                                  const float* __restrict__ vheap,
                                  const int*  __restrict__ block_table,
                                  const int*  __restrict__ ctx_lens,
                                  float* __restrict__ attn_out)
{
    const int b   = blockIdx.x;
    const int h   = blockIdx.y;
    const int tid = threadIdx.x;               // 256 threads

    __shared__ float qs[HD];
    __shared__ float sc[MAXS];
    __shared__ float red[8];
    __shared__ float acc_sh[HD];

    const int   ctx   = ctx_lens[b];
    const float scale = 0.088388347648318447f;  // 1/sqrt(128)

    if (tid < HD) qs[tid] = q[(size_t)b * DIM + h * HD + tid] * scale;
    __syncthreads();

    // ---- phase 1: scores.  8-lane group per position (512B coalesced/pos)
    const int grp = tid >> 3;                  // 0..31
    const int gl  = tid & 7;
    for (int s = grp; s < ctx; s += 32) {
        const float* krow;
        if (s == ctx - 1) {
            krow = knew + (size_t)b * DIM + h * HD;
        } else {
            const int blk  = block_table[b * BLK_PER + (s >> 4)];
            const int slot = blk * BS + (s & 15);
            krow = kheap + (size_t)slot * (NH * HD) + h * HD;
        }
        float p = 0.f;
#pragma unroll
        for (int r = 0; r < 4; ++r) {
            const int d = r * 32 + gl * 4;     // group covers 128B per round
            const v4f kv = __builtin_nontemporal_load((const v4f*)(krow + d));
            p += kv.x * qs[d] + kv.y * qs[d + 1] + kv.z * qs[d + 2] + kv.w * qs[d + 3];
        }
        p += __shfl_xor(p, 1, 32);
        p += __shfl_xor(p, 2, 32);
        p += __shfl_xor(p, 4, 32);
        if (gl == 0) sc[s] = p;
    }
    __syncthreads();

    // ---- softmax over sc[0..ctx)
    float m = -1e30f;
    for (int s = tid; s < ctx; s += 256) m = fmaxf(m, sc[s]);
    for (int off = 16; off > 0; off >>= 1) m = fmaxf(m, __shfl_xor(m, off, 32));
    if ((tid & 31) == 0) red[tid >> 5] = m;
    __syncthreads();
    m = red[0];
#pragma unroll
    for (int i = 1; i < 8; ++i) m = fmaxf(m, red[i]);
    __syncthreads();

    float ls = 0.f;
    for (int s = tid; s < ctx; s += 256) {
        const float e = __expf(sc[s] - m);
        sc[s] = e;
        ls += e;
    }
    for (int off = 16; off > 0; off >>= 1) ls += __shfl_xor(ls, off, 32);
    if ((tid & 31) == 0) red[tid >> 5] = ls;
    __syncthreads();
    float sum = 0.f;
#pragma unroll
    for (int i = 0; i < 8; ++i) sum += red[i];
    const float inv = 1.0f / sum;

    // ---- phase 2: out[d] = sum_s p[s] * V[s][d]   (512B coalesced per s)
    const int d    = tid & (HD - 1);
    const int hlf  = tid >> 7;                 // 0/1: split positions
    float acc = 0.f;
    for (int s = hlf; s < ctx; s += 2) {
        const float* vrow;
        if (s == ctx - 1) {
            vrow = vnew + (size_t)b * DIM + h * HD;
        } else {
            const int blk  = block_table[b * BLK_PER + (s >> 4)];
            const int slot = blk * BS + (s & 15);
            vrow = vheap + (size_t)slot * (NH * HD) + h * HD;
        }
        acc += sc[s] * __builtin_nontemporal_load(vrow + d);
    }
    if (hlf == 0) acc_sh[d] = acc;
    __syncthreads();
    if (hlf == 1)
        attn_out[(size_t)b * DIM + h * HD + d] = (acc_sh[d] + acc) * inv;
}

// ---------------------------------------------------------------------------
// Host-side orchestration
// ---------------------------------------------------------------------------
extern "C" void kernel_launch(void* const* d_in, const int* in_sizes, int n_in,
                              void* d_out, int out_size, void* d_ws, size_t ws_size,
                              hipStream_t stream)
{
    (void)in_sizes; (void)n_in; (void)out_size; (void)ws_size;

    const float* x        = (const float*)d_in[0];
    const float* kheap    = (const float*)d_in[1];
    const float* vheap    = (const float*)d_in[2];
    const float* wq       = (const float*)d_in[3];
    const float* wk       = (const float*)d_in[4];
    const float* wv       = (const float*)d_in[5];
    const float* wo       = (const float*)d_in[6];
    const float* w1       = (const float*)d_in[7];
    const float* w2       = (const float*)d_in[8];
    const float* w3       = (const float*)d_in[9];
    const float* norm1_w  = (const float*)d_in[10];
    const float* norm2_w  = (const float*)d_in[11];
    const int*   btab     = (const int*)d_in[12];
    // d_in[13] slot_mapping: implied by ctx-1 substitution
    const int*   ctx_lens = (const int*)d_in[14];
    float*       out      = (float*)d_out;

    // workspace layout (floats) -- ~7.1 MB total
    float* ws   = (float*)d_ws;
    float* h1   = ws;                     // 32*2048
    float* qb   = h1   + BATCH * DIM;
    float* kn   = qb   + BATCH * DIM;
    float* vn   = kn   + BATCH * DIM;
    float* attn = vn   + BATCH * DIM;
    float* x2   = attn + BATCH * DIM;
    float* h2   = x2   + BATCH * DIM;
    float* gb   = h2   + BATCH * DIM;     // 32*8192
    float* part = gb   + BATCH * HID;     // up to 2*2*32*8192 = 1,048,576 floats

    const int nD = BATCH * DIM;           // 65536
    const int nH = BATCH * HID;           // 262144

    // 1) h1 = rmsnorm(x)
    rmsnorm_kernel<<<BATCH, 256, 0, stream>>>(x, norm1_w, h1);

    // 2) fused QKV, split-K S=4 (384 WGs): q=h1@wq^T, k=h1@wk^T, v=h1@wv^T
    gemm32_part_kernel<<<dim3(DIM / 64, 3, 4), 256, 0, stream>>>(
        h1, wq, wk, wv, part, DIM, DIM, DIM / 4);
    reduce_kernel<<<dim3((nD + 255) / 256, 3), 256, 0, stream>>>(
        part, 4, nD, qb, kn, vn, nullptr);

    // 3) paged attention (decode position folded in as s == ctx-1)
    paged_attn_kernel<<<dim3(BATCH, NH), 256, 0, stream>>>(
        qb, kn, vn, kheap, vheap, btab, ctx_lens, attn);

    // 4) x2 = attn @ wo^T + x, split-K S=8 (256 WGs)
    gemm32_part_kernel<<<dim3(DIM / 64, 1, 8), 256, 0, stream>>>(
        attn, wo, nullptr, nullptr, part, DIM, DIM, DIM / 8);
    reduce_kernel<<<dim3((nD + 255) / 256, 1), 256, 0, stream>>>(
        part, 8, nD, x2, nullptr, nullptr, x);

    // 5) h2 = rmsnorm(x2)
    rmsnorm_kernel<<<BATCH, 256, 0, stream>>>(x2, norm2_w, h2);

    // 6) fused gate/up, split-K S=2 (512 WGs): a=h2@w1^T, c=h2@w3^T
    gemm32_part_kernel<<<dim3(HID / 64, 2, 2), 256, 0, stream>>>(
        h2, w1, w3, nullptr, part, DIM, HID, DIM / 2);
    //    fused reduce + SwiGLU: g = silu(a) * c
    swiglu_reduce_kernel<<<(nH + 255) / 256, 256, 0, stream>>>(part, 2, nH, gb);

    // 7) out = g @ w2^T + x2, split-K S=8 (256 WGs)
    gemm32_part_kernel<<<dim3(DIM / 64, 1, 8), 256, 0, stream>>>(
        gb, w2, nullptr, nullptr, part, HID, DIM, HID / 8);
    reduce_kernel<<<dim3((nD + 255) / 256, 1), 256, 0, stream>>>(
        part, 8, nD, out, nullptr, nullptr, x2);
}